// RotSSM_3685081940233
// MI455X (gfx1250) — compile-verified
//
#include <hip/hip_runtime.h>
#include <hip/hip_bf16.h>

typedef __attribute__((ext_vector_type(2))) float v2f;
typedef __attribute__((ext_vector_type(8))) float v8f;

#define HH      64       // heads
#define NN      4        // state per head
#define DH      256      // hidden
#define SD      256      // total state dim (HH*NN)
#define TT      4096
#define BSZ     8
#define LCHUNK  64
#define NCHUNK  (TT / LCHUNK)   // 64
#define LDSW    (LCHUNK + 1)    // 65, conflict-free stride

// ---------------------------------------------------------------------------
// Kernel 0: per-head precompute.  gamma, normalization, P = expm(skew) via
// scaled Taylor + squaring, folded weights W1 = P*B_norm (per-head rows) and
// W2 = C * blockdiag(P^T), plus rotation/decay tables (per-step and per-chunk).
// ---------------------------------------------------------------------------
__global__ void rotssm_precompute(
    const float* __restrict__ thetas_log, const float* __restrict__ P_param,
    const float* __restrict__ B_param, const float* __restrict__ C,
    const float* __restrict__ gamma_log,
    float* __restrict__ W1, float* __restrict__ W2,
    float* __restrict__ cst, float* __restrict__ snt, float* __restrict__ gst,
    float* __restrict__ clT, float* __restrict__ slT, float* __restrict__ glT)
{
    int h = threadIdx.x;
    if (h >= HH) return;

    float gamma = __expf(-__expf(gamma_log[h]));

    const float* Bp = B_param + (size_t)h * NN * DH;
    float tr = 0.f;
    for (int i = 0; i < NN * DH; ++i) { float v = Bp[i]; tr += v * v; }
    float nrm = sqrtf((1.f - gamma * gamma) / tr);

    // skew = Pp - Pp^T, scaled by 2^-6; expm via 12-term Taylor then 6 squarings
    float A[16], X[16], Pm[16], Tm[16];
    for (int r = 0; r < 4; ++r)
        for (int c = 0; c < 4; ++c)
            A[r*4+c] = (P_param[h*16 + r*4 + c] - P_param[h*16 + c*4 + r]) * (1.0f / 64.0f);
    for (int i = 0; i < 16; ++i) { Pm[i] = (i % 5 == 0) ? 1.f : 0.f; X[i] = Pm[i]; }
    for (int k = 1; k <= 12; ++k) {
        for (int r = 0; r < 4; ++r)
            for (int c = 0; c < 4; ++c) {
                float s = 0.f;
                for (int j = 0; j < 4; ++j) s += X[r*4+j] * A[j*4+c];
                Tm[r*4+c] = s;
            }
        float inv = 1.0f / (float)k;
        for (int i = 0; i < 16; ++i) { X[i] = Tm[i] * inv; Pm[i] += X[i]; }
    }
    for (int sq = 0; sq < 6; ++sq) {
        for (int r = 0; r < 4; ++r)
            for (int c = 0; c < 4; ++c) {
                float s = 0.f;
                for (int j = 0; j < 4; ++j) s += Pm[r*4+j] * Pm[j*4+c];
                Tm[r*4+c] = s;
            }
        for (int i = 0; i < 16; ++i) Pm[i] = Tm[i];
    }

    // W1[(h*4+N)][D] = nrm * sum_n P[N][n] * B_param[h][n][D]
    for (int D = 0; D < DH; ++D) {
        for (int N = 0; N < 4; ++N) {
            float s = 0.f;
            for (int n = 0; n < 4; ++n) s += Pm[N*4+n] * Bp[n*DH + D];
            W1[(size_t)(h*4 + N) * DH + D] = nrm * s;
        }
    }
    // W2[Dr][(h*4+N)] = sum_n C[Dr][h*4+n] * P[N][n]
    for (int Dr = 0; Dr < DH; ++Dr) {
        for (int N = 0; N < 4; ++N) {
            float s = 0.f;
            for (int n = 0; n < 4; ++n) s += C[(size_t)Dr * SD + h*4 + n] * Pm[N*4+n];
            W2[(size_t)Dr * SD + h*4 + N] = s;
        }
    }

    // rotation/decay tables (per state index), per-step and per-chunk (A^L)
    float gL = __expf((float)LCHUNK * __logf(gamma));
    for (int p = 0; p < 2; ++p) {
        float th = __expf(thetas_log[h*2 + p]);
        float c, s, cL, sL;
        __sincosf(th, &s, &c);
        __sincosf(th * (float)LCHUNK, &sL, &cL);
        for (int q = 0; q < 2; ++q) {
            int i = h*4 + p*2 + q;
            cst[i] = c;  snt[i] = s;  gst[i] = gamma;
            clT[i] = cL; slT[i] = sL; glT[i] = gL;
        }
    }
}

// ---------------------------------------------------------------------------
// Pass 1: per (batch, chunk): GEMM1 (WMMA f32 16x16x4) -> local scan (zero
// initial state) -> GEMM2 + D*u, writing partial output + local end-state.
// ---------------------------------------------------------------------------
__global__ __launch_bounds__(256)
void rotssm_chunk(const float* __restrict__ u,
                  const float* __restrict__ W1, const float* __restrict__ W2,
                  const float* __restrict__ Dv,
                  const float* __restrict__ cst, const float* __restrict__ snt,
                  const float* __restrict__ gst,
                  float* __restrict__ out, float* __restrict__ Slocal)
{
    __shared__ float uT[SD * LDSW];   // u tile, [D][t]
    __shared__ float ys[SD * LDSW];   // Us -> local scan result, [state][t]

    const int b     = blockIdx.y;
    const int chunk = blockIdx.x;
    const int c0    = chunk * LCHUNK;
    const int tid   = threadIdx.x;
    const int lane  = tid & 31;
    const int wv    = tid >> 5;       // 8 waves
    const int half  = lane >> 4;
    const int l16   = lane & 15;
    const int mB    = wv * 32;        // 32-row M strip per wave

    // ---- stage u tile (coalesced float4) ----
    {
        const float* ub = u + (size_t)b * DH * TT + c0;
        const int tq = (tid & 15) * 4;
        for (int Dr = tid >> 4; Dr < DH; Dr += 16) {
            const float4 v = *(const float4*)(ub + (size_t)Dr * TT + tq);
            float* dst = &uT[Dr * LDSW + tq];
            dst[0] = v.x; dst[1] = v.y; dst[2] = v.z; dst[3] = v.w;
        }
    }
    __syncthreads();

    // ---- GEMM1: ys = W1 (256x256) * uT (256x64) ----
    v8f acc[2][4];
    for (int i = 0; i < 2; ++i)
        for (int j = 0; j < 4; ++j) acc[i][j] = (v8f){};
    {
        const float* W1r0 = W1 + (size_t)(mB + l16) * DH;
        const float* W1r1 = W1 + (size_t)(mB + 16 + l16) * DH;
        for (int k = 0; k < DH; k += 4) {
            const int ko = k + 2 * half;
            v2f a0 = *(const v2f*)(W1r0 + ko);
            v2f a1 = *(const v2f*)(W1r1 + ko);
#pragma unroll
            for (int j = 0; j < 4; ++j) {
                const int col = j * 16 + l16;
                v2f bf;
                bf.x = uT[ko * LDSW + col];
                bf.y = uT[(ko + 1) * LDSW + col];
                acc[0][j] = __builtin_amdgcn_wmma_f32_16x16x4_f32(
                    false, a0, false, bf, (short)0, acc[0][j], false, false);
                acc[1][j] = __builtin_amdgcn_wmma_f32_16x16x4_f32(
                    false, a1, false, bf, (short)0, acc[1][j], false, false);
            }
        }
    }
#pragma unroll
    for (int i = 0; i < 2; ++i)
        for (int j = 0; j < 4; ++j) {
            const int rowb = mB + i * 16 + 8 * half;
            const int col  = j * 16 + l16;
#pragma unroll
            for (int e = 0; e < 8; ++e) ys[(rowb + e) * LDSW + col] = acc[i][j][e];
        }
    __syncthreads();

    // ---- local scan over time (state idx == tid; pair partner via shfl_xor) --
    {
        const int i = tid;
        const float c  = cst[i];
        const float s  = snt[i];
        const float g  = gst[i];
        const float ss = (i & 1) ? s : -s;   // even: c*x - s*xp ; odd: c*x + s*xp
        float x = 0.f;
        float* row = &ys[i * LDSW];
        for (int t = 0; t < LCHUNK; ++t) {
            float xp = __shfl_xor(x, 1, 32);
            x = g * (c * x + ss * xp) + row[t];
            row[t] = x;
        }
        Slocal[(size_t)(b * NCHUNK + chunk) * SD + i] = x;
    }
    __syncthreads();

    // ---- GEMM2: out = W2 (256x256) * ys (256x64) + D*u ----
    v8f acc2[2][4];
    for (int i = 0; i < 2; ++i)
        for (int j = 0; j < 4; ++j) acc2[i][j] = (v8f){};
    {
        const float* W2r0 = W2 + (size_t)(mB + l16) * SD;
        const float* W2r1 = W2 + (size_t)(mB + 16 + l16) * SD;
        for (int k = 0; k < SD; k += 4) {
            const int ko = k + 2 * half;
            v2f a0 = *(const v2f*)(W2r0 + ko);
            v2f a1 = *(const v2f*)(W2r1 + ko);
#pragma unroll
            for (int j = 0; j < 4; ++j) {
                const int col = j * 16 + l16;
                v2f bf;
                bf.x = ys[ko * LDSW + col];
                bf.y = ys[(ko + 1) * LDSW + col];
                acc2[0][j] = __builtin_amdgcn_wmma_f32_16x16x4_f32(
                    false, a0, false, bf, (short)0, acc2[0][j], false, false);
                acc2[1][j] = __builtin_amdgcn_wmma_f32_16x16x4_f32(
                    false, a1, false, bf, (short)0, acc2[1][j], false, false);
            }
        }
    }
    {
        float* ob = out + (size_t)b * DH * TT + c0;
#pragma unroll
        for (int i = 0; i < 2; ++i)
            for (int j = 0; j < 4; ++j) {
                const int rowb = mB + i * 16 + 8 * half;
                const int col  = j * 16 + l16;
#pragma unroll
                for (int e = 0; e < 8; ++e) {
                    const int row = rowb + e;
                    ob[(size_t)row * TT + col] =
                        acc2[i][j][e] + Dv[row] * uT[row * LDSW + col];
                }
            }
    }
}

// ---------------------------------------------------------------------------
// Pass 2: sequential prefix over chunks using the closed-form chunk operator
// A^L = gamma^L * R(L*theta).  One thread per (batch, head).
// Eprev[b][k] = true end-state of chunk k-1 (zero for k==0).
// ---------------------------------------------------------------------------
__global__ void rotssm_prefix(const float* __restrict__ Slocal,
                              const float* __restrict__ clT,
                              const float* __restrict__ slT,
                              const float* __restrict__ glT,
                              float* __restrict__ Eprev)
{
    const int b = blockIdx.x;
    const int h = threadIdx.x;
    if (h >= HH) return;
    const int i0 = h * 4;
    const float cl0 = clT[i0],     sl0 = slT[i0];
    const float cl1 = clT[i0 + 2], sl1 = slT[i0 + 2];
    const float gl  = glT[i0];
    float e0 = 0.f, e1 = 0.f, e2 = 0.f, e3 = 0.f;
    for (int k = 0; k < NCHUNK; ++k) {
        float* ep = Eprev + (size_t)(b * NCHUNK + k) * SD + i0;
        ep[0] = e0; ep[1] = e1; ep[2] = e2; ep[3] = e3;
        const float* sp = Slocal + (size_t)(b * NCHUNK + k) * SD + i0;
        const float n0 = gl * (cl0 * e0 - sl0 * e1) + sp[0];
        const float n1 = gl * (cl0 * e1 + sl0 * e0) + sp[1];
        const float n2 = gl * (cl1 * e2 - sl1 * e3) + sp[2];
        const float n3 = gl * (cl1 * e3 + sl1 * e2) + sp[3];
        e0 = n0; e1 = n1; e2 = n2; e3 = n3;
    }
}

// ---------------------------------------------------------------------------
// Pass 3: per (batch, chunk>=1): build correction columns A^{dt+1}*E_prev in
// LDS, then out += W2 * correction (WMMA GEMM).
// ---------------------------------------------------------------------------
__global__ __launch_bounds__(256)
void rotssm_correct(const float* __restrict__ W2,
                    const float* __restrict__ Eprev,
                    const float* __restrict__ cst, const float* __restrict__ snt,
                    const float* __restrict__ gst,
                    float* __restrict__ out)
{
    __shared__ float Mc[SD * LDSW];

    const int b     = blockIdx.y;
    const int chunk = blockIdx.x + 1;      // chunk 0 needs no correction
    const int c0    = chunk * LCHUNK;
    const int tid   = threadIdx.x;
    const int lane  = tid & 31;
    const int wv    = tid >> 5;
    const int half  = lane >> 4;
    const int l16   = lane & 15;
    const int mB    = wv * 32;

    if (tid < 128) {                       // one thread per rotation pair
        const int i0 = 2 * tid;
        const float c = cst[i0], s = snt[i0], g = gst[i0];
        const float* ep = Eprev + (size_t)(b * NCHUNK + chunk) * SD + i0;
        float v0 = ep[0], v1 = ep[1];
        for (int t = 0; t < LCHUNK; ++t) {
            const float n0 = g * (c * v0 - s * v1);
            const float n1 = g * (c * v1 + s * v0);
            v0 = n0; v1 = n1;
            Mc[i0 * LDSW + t]       = v0;
            Mc[(i0 + 1) * LDSW + t] = v1;
        }
    }
    __syncthreads();

    v8f acc[2][4];
    for (int i = 0; i < 2; ++i)
        for (int j = 0; j < 4; ++j) acc[i][j] = (v8f){};
    {
        const float* W2r0 = W2 + (size_t)(mB + l16) * SD;
        const float* W2r1 = W2 + (size_t)(mB + 16 + l16) * SD;
        for (int k = 0; k < SD; k += 4) {
            const int ko = k + 2 * half;
            v2f a0 = *(const v2f*)(W2r0 + ko);
            v2f a1 = *(const v2f*)(W2r1 + ko);
#pragma unroll
            for (int j = 0; j < 4; ++j) {
                const int col = j * 16 + l16;
                v2f bf;
                bf.x = Mc[ko * LDSW + col];
                bf.y = Mc[(ko + 1) * LDSW + col];
                acc[0][j] = __builtin_amdgcn_wmma_f32_16x16x4_f32(
                    false, a0, false, bf, (short)0, acc[0][j], false, false);
                acc[1][j] = __builtin_amdgcn_wmma_f32_16x16x4_f32(
                    false, a1, false, bf, (short)0, acc[1][j], false, false);
            }
        }
    }
    {
        float* ob = out + (size_t)b * DH * TT + c0;
#pragma unroll
        for (int i = 0; i < 2; ++i)
            for (int j = 0; j < 4; ++j) {
                const int rowb = mB + i * 16 + 8 * half;
                const int col  = j * 16 + l16;
#pragma unroll
                for (int e = 0; e < 8; ++e) {
                    const int row = rowb + e;
                    const size_t idx = (size_t)row * TT + col;
                    ob[idx] = ob[idx] + acc[i][j][e];
                }
            }
    }
}

// ---------------------------------------------------------------------------
extern "C" void kernel_launch(void* const* d_in, const int* in_sizes, int n_in,
                              void* d_out, int out_size, void* d_ws, size_t ws_size,
                              hipStream_t stream) {
    const float* u          = (const float*)d_in[0];
    const float* thetas_log = (const float*)d_in[1];
    const float* P_param    = (const float*)d_in[2];
    const float* B_param    = (const float*)d_in[3];
    const float* C          = (const float*)d_in[4];
    const float* Dv         = (const float*)d_in[5];
    const float* gamma_log  = (const float*)d_in[6];
    float* out = (float*)d_out;

    float* ws = (float*)d_ws;
    float* W1  = ws;                         // 256*256
    float* W2  = W1  + SD * DH;              // 256*256
    float* cst = W2  + SD * DH;              // 256
    float* snt = cst + SD;
    float* gst = snt + SD;
    float* clT = gst + SD;
    float* slT = clT + SD;
    float* glT = slT + SD;
    float* Slocal = glT + SD;                // B*NCHUNK*256
    float* Eprev  = Slocal + (size_t)BSZ * NCHUNK * SD;

    rotssm_precompute<<<1, 64, 0, stream>>>(thetas_log, P_param, B_param, C,
                                            gamma_log, W1, W2, cst, snt, gst,
                                            clT, slT, glT);

    rotssm_chunk<<<dim3(NCHUNK, BSZ), 256, 0, stream>>>(
        u, W1, W2, Dv, cst, snt, gst, out, Slocal);

    rotssm_prefix<<<BSZ, 64, 0, stream>>>(Slocal, clT, slT, glT, Eprev);

    rotssm_correct<<<dim3(NCHUNK - 1, BSZ), 256, 0, stream>>>(
        W2, Eprev, cst, snt, gst, out);
}